// ModelParallelStage_18141941859023
// MI455X (gfx1250) — compile-verified
//
#include <hip/hip_runtime.h>
#include <math.h>

// Problem constants (match reference setup_inputs)
#define NW 4         // workers
#define NN 100000    // nodes
#define NF 128       // features == hidden
#define NE 1600000   // edges

typedef float v2f __attribute__((ext_vector_type(2)));
typedef float v8f __attribute__((ext_vector_type(8)));

// ---------------------------------------------------------------------------
// Kernel 1: zero the agg buffer (== d_out, used as scatter target) and the
// degree scratch in d_ws.  d_out is poisoned once and never re-poisoned, so
// this must run every launch.
// ---------------------------------------------------------------------------
__global__ void zero_kernel(float* __restrict__ out, size_t nout,
                            float* __restrict__ ws, size_t nws) {
    size_t stride = (size_t)gridDim.x * blockDim.x;
    size_t start  = (size_t)blockIdx.x * blockDim.x + threadIdx.x;
    for (size_t i = start; i < nout; i += stride) out[i] = 0.0f;
    for (size_t i = start; i < nws;  i += stride) ws[i]  = 0.0f;
}

// ---------------------------------------------------------------------------
// Kernel 2: degree histograms via f32 atomics (counts are small integers,
// exactly representable).  One thread per (worker, edge).
// ---------------------------------------------------------------------------
__global__ void degree_kernel(const long long* __restrict__ src,
                              const long long* __restrict__ dst,
                              float* __restrict__ outdeg,
                              float* __restrict__ indeg) {
    size_t idx = (size_t)blockIdx.x * blockDim.x + threadIdx.x;
    if (idx >= (size_t)NW * NE) return;
    int w = (int)(idx / NE);
    int s = (int)src[idx];
    int d = (int)dst[idx];
    atomicAdd(outdeg + (size_t)w * NN + s, 1.0f);
    atomicAdd(indeg  + (size_t)w * NN + d, 1.0f);
}

// ---------------------------------------------------------------------------
// Kernel 3: deg -> clip(deg,1)^-0.5, in place over both degree arrays.
// ---------------------------------------------------------------------------
__global__ void scale_kernel(float* __restrict__ deg, int n) {
    int i = blockIdx.x * blockDim.x + threadIdx.x;
    if (i < n) deg[i] = 1.0f / sqrtf(fmaxf(deg[i], 1.0f));
}

// ---------------------------------------------------------------------------
// Kernel 4: edge scatter:  agg[w][dst] += feats[w][src] * outscale[w][src].
// One wave32 per edge; lane l handles features [4l, 4l+4): one coalesced
// 16-byte row load + 4 global_atomic_add_f32.  Memory-bound phase.
// ---------------------------------------------------------------------------
__global__ void scatter_kernel(const float* __restrict__ feats,
                               const long long* __restrict__ src,
                               const long long* __restrict__ dst,
                               const float* __restrict__ oscale,
                               float* __restrict__ agg) {
    size_t gid  = (size_t)blockIdx.x * blockDim.x + threadIdx.x;
    size_t edge = gid >> 5;
    int    lane = (int)(gid & 31);
    if (edge >= (size_t)NW * NE) return;
    int w = (int)(edge / NE);
    int s = (int)src[edge];
    int d = (int)dst[edge];
    float sc = oscale[(size_t)w * NN + s];
    const float4* frow = (const float4*)(feats + ((size_t)w * NN + s) * NF);
    float4 v = frow[lane];
    float* arow = agg + ((size_t)w * NN + d) * NF + lane * 4;
    atomicAdd(arow + 0, v.x * sc);
    atomicAdd(arow + 1, v.y * sc);
    atomicAdd(arow + 2, v.z * sc);
    atomicAdd(arow + 3, v.w * sc);
}

// ---------------------------------------------------------------------------
// Kernel 5: out[w][n][:] = (agg[w][n][:] * inscale[w][n]) @ W + bias
// In-place on d_out.  8 waves / block, each wave owns a 16-row M-tile and all
// 8 N-tiles, iterating K in 32 steps of 4 with V_WMMA_F32_16X16X4_F32 (fp32
// precision preserved).  Weight (64 KB) is staged in LDS pre-swizzled as
// (K-pair, N) float2 so each B fragment is one ds_load_b64.
//
// Fragment layouts (ISA 7.12.2):
//   A 16x4 f32 : lane<16 -> K={k0,k0+1}, lane>=16 -> K={k0+2,k0+3}, M=lane&15
//   B 4x16 f32 : mirrored per-lane K pattern, N = lane&15
//   C/D 16x16  : VGPR v -> M = v + 8*(lane>>4), N = lane&15
// ---------------------------------------------------------------------------
__global__ void __launch_bounds__(256)
gemm_kernel(float* __restrict__ inout,          // agg in, result out (in-place)
            const float* __restrict__ weight,   // [128,128] row-major (K,N)
            const float* __restrict__ bias,     // [128]
            const float* __restrict__ iscale) { // [NW*NN]
    __shared__ float2 lwt[64 * 128];            // 64 KB: [kpair][n] = {W[2p][n], W[2p+1][n]}

    int tid = threadIdx.x;
    for (int i = tid; i < 64 * 128; i += 256) {
        int p = i >> 7;
        int n = i & 127;
        lwt[i] = make_float2(weight[(2 * p) * NF + n], weight[(2 * p + 1) * NF + n]);
    }
    __syncthreads();

    int wave = tid >> 5;
    int lane = tid & 31;
    int tile = blockIdx.x * 8 + wave;           // exact grid: no divergence
    int w    = tile / (NN / 16);
    int row0 = (tile % (NN / 16)) * 16;
    int half = lane >> 4;                        // 0 or 1
    int m    = lane & 15;

    size_t rowbase = ((size_t)w * NN + row0 + m) * NF;
    float  rscale  = iscale[(size_t)w * NN + row0 + m]; // in-deg^-0.5 for this A row

    v8f c[8];
#pragma unroll
    for (int nt = 0; nt < 8; ++nt) c[nt] = (v8f)(0.0f);

    const float2* arow = (const float2*)(inout + rowbase);
#pragma unroll 4
    for (int ks = 0; ks < 32; ++ks) {
        int kp = ks * 2 + half;                  // K-pair index this lane holds
        float2 af = arow[kp];
        v2f a;
        a.x = af.x * rscale;
        a.y = af.y * rscale;
#pragma unroll
        for (int nt = 0; nt < 8; ++nt) {
            float2 bf = lwt[kp * 128 + nt * 16 + m];
            v2f b;
            b.x = bf.x;
            b.y = bf.y;
            c[nt] = __builtin_amdgcn_wmma_f32_16x16x4_f32(
                /*neg_a=*/false, a, /*neg_b=*/false, b,
                /*c_mod=*/(short)0, c[nt], /*reuse_a=*/false, /*reuse_b=*/false);
        }
    }

    // Write back with bias (all A reads for this tile are complete; rows are
    // wave-private, so in-place overwrite of agg with the result is safe).
#pragma unroll
    for (int nt = 0; nt < 8; ++nt) {
        int   n  = nt * 16 + m;
        float bv = bias[n];
#pragma unroll
        for (int v = 0; v < 8; ++v) {
            inout[((size_t)w * NN + row0 + v + 8 * half) * NF + n] = c[nt][v] + bv;
        }
    }
}

// ---------------------------------------------------------------------------
extern "C" void kernel_launch(void* const* d_in, const int* in_sizes, int n_in,
                              void* d_out, int out_size, void* d_ws, size_t ws_size,
                              hipStream_t stream) {
    const float*     feats  = (const float*)d_in[0];      // [NW,NN,NF] f32
    const float*     weight = (const float*)d_in[1];      // [NF,NF]    f32
    const float*     bias   = (const float*)d_in[2];      // [NF]       f32
    const long long* src    = (const long long*)d_in[3];  // [NW,NE]    i64
    const long long* dst    = (const long long*)d_in[4];  // [NW,NE]    i64
    float* out = (float*)d_out;                           // [NW,NN,NF] f32 (also agg)

    float* oscale = (float*)d_ws;                 // NW*NN floats
    float* iscale = oscale + (size_t)NW * NN;     // NW*NN floats  (3.2 MB total ws)

    // 1) zero agg (=d_out) and both degree arrays
    zero_kernel<<<2048, 256, 0, stream>>>(out, (size_t)NW * NN * NF,
                                          oscale, (size_t)2 * NW * NN);
    // 2) degree histograms
    degree_kernel<<<(NW * NE + 255) / 256, 256, 0, stream>>>(src, dst, oscale, iscale);
    // 3) clip+rsqrt both arrays in place
    scale_kernel<<<(2 * NW * NN + 255) / 256, 256, 0, stream>>>(oscale, 2 * NW * NN);
    // 4) scatter-add (wave per edge): 6.4M edges * 32 lanes / 256 = 800000 blocks
    scatter_kernel<<<(int)(((size_t)NW * NE * 32) / 256), 256, 0, stream>>>(
        feats, src, dst, oscale, out);
    // 5) in-place WMMA GEMM + in-scale + bias: 25000 tiles / 8 waves = 3125 blocks
    gemm_kernel<<<(NW * (NN / 16)) / 8, 256, 0, stream>>>(out, weight, bias, iscale);
}